// PPScatter_25924422599253
// MI455X (gfx1250) — compile-verified
//
#include <hip/hip_runtime.h>
#include <stdint.h>

// Problem constants from the reference.
#define B_  8
#define C_  64
#define H_  496
#define W_  432
#define HW_ (H_ * W_)
#define TILE 256   // pillars per block; LDS tile = 64 * 256 * 4B = 64 KB

#ifdef __has_builtin
#if __has_builtin(__builtin_amdgcn_tensor_load_to_lds)
#define HAVE_TDM 1
#endif
#if __has_builtin(__builtin_amdgcn_s_wait_tensorcnt)
#define HAVE_TCNT 1
#endif
#endif

typedef __attribute__((ext_vector_type(4))) unsigned int u32x4;
typedef __attribute__((ext_vector_type(4))) int          i32x4;
typedef __attribute__((ext_vector_type(8))) int          i32x8;

__device__ __forceinline__ void wait_tensorcnt0() {
#ifdef HAVE_TCNT
    __builtin_amdgcn_s_wait_tensorcnt(0);
#else
    asm volatile("s_wait_tensorcnt 0" ::: "memory");
#endif
}

// ---------------------------------------------------------------------------
// Kernel 1: zero the 439 MB output with 128-bit stores (pure bandwidth).
// ---------------------------------------------------------------------------
__global__ void PPScatter_zero_kernel(float4* __restrict__ out4, int n4,
                                      float* __restrict__ out_tail, int tail) {
    int i = blockIdx.x * blockDim.x + threadIdx.x;
    const float4 z = make_float4(0.f, 0.f, 0.f, 0.f);
    if (i < n4) out4[i] = z;
    if (i == 0) {
        for (int t = 0; t < tail; ++t) out_tail[t] = 0.f;
    }
}

// ---------------------------------------------------------------------------
// Kernel 2: TDM-staged pillar scatter.
//   Block = (batch b, pillar range [p0, p0+rem)).
//   Wave 0 issues one tensor_load_to_lds pulling the 2D tile
//   x[b, 0..63, p0..p0+rem)  (row stride P)  into LDS, then everyone scatters.
// ---------------------------------------------------------------------------
__global__ __launch_bounds__(TILE) void
PPScatter_scatter_kernel(const float* __restrict__ x,
                         const int*   __restrict__ inds,
                         float*       __restrict__ out,
                         int P, int blocksPerBatch) {
    __shared__ float tile[C_ * TILE];

    const int b  = blockIdx.x / blocksPerBatch;
    const int t  = blockIdx.x - b * blocksPerBatch;
    const int p0 = t * TILE;
    int rem = P - p0;
    if (rem > TILE) rem = TILE;
    if (rem <= 0) return;

    const float* src = x + ((size_t)b * C_) * (size_t)P + (size_t)p0;

#ifdef HAVE_TDM
    // One wave builds the Tensor DMA descriptor (D#) and issues the TDM load.
    if (threadIdx.x < 32) {
        const uint64_t ga      = (uint64_t)(uintptr_t)src;
        const uint32_t lds_off = (uint32_t)(uintptr_t)(&tile[0]); // low 32b of flat LDS ptr = LDS byte offset

        // ---- D# group 0 (128b): control | lds_addr | global_addr | type ----
        u32x4 g0;
        g0[0] = 1u;                                    // count=1, is_restore=0, gather off
        g0[1] = lds_off;                               // lds_addr (bytes)
        g0[2] = (uint32_t)(ga & 0xFFFFFFFFu);          // global_addr[31:0]
        g0[3] = (uint32_t)((ga >> 32) & 0x01FFFFFFu)   // global_addr[56:32]
              | (2u << 30);                            // type = 2 ("image")

        // ---- D# group 1 (256b): sizes / strides ----
        const uint32_t dim0 = (uint32_t)P;             // tensor_dim0 (elements)
        i32x8 g1;
        g1[0] = (int)(2u << 16);                       // workgroup_mask=0, data_size=2 (4B), no pad/iterate
        g1[1] = (int)((dim0 & 0xFFFFu) << 16);         // atomic_barrier_addr=0 | tensor_dim0[15:0]
        g1[2] = (int)(((dim0 >> 16) & 0xFFFFu)         // tensor_dim0[31:16]
              | ((uint32_t)C_ << 16));                 // tensor_dim1[15:0] = 64
        g1[3] = (int)((uint32_t)rem << 16);            // tensor_dim1[31:16]=0 | tile_dim0 = rem
        g1[4] = (int)C_;                               // tile_dim1 = 64, tile_dim2 = 0
        g1[5] = (int)dim0;                             // tensor_dim0_stride[31:0] = P
        g1[6] = 0;                                     // stride[47:32]=0 | tensor_dim1_stride lo = 0
        g1[7] = 0;                                     // tensor_dim1_stride hi = 0

        const i32x4 gz4 = {0, 0, 0, 0};                // groups 2/3 unused (2D tile)
        const i32x8 gz8 = {0, 0, 0, 0, 0, 0, 0, 0};    // extra group (6-arg form), unused
        __builtin_amdgcn_tensor_load_to_lds(g0, g1, gz4, gz4, gz8, 0);
        wait_tensorcnt0();
    }
    __syncthreads();  // LDS tile now holds x[b, c, p0+j] at tile[c*rem + j]
#endif

    const int tid = threadIdx.x;
    if (tid < rem) {
        const int p = p0 + tid;
        const int* ip = inds + ((size_t)b * (size_t)P + (size_t)p) * 3;
        const int m  = ip[0];
        const int xi = ip[1];
        const int yi = ip[2];
        if (m == 1) {
            const size_t obase = (size_t)b * (size_t)(C_ * HW_)
                               + (size_t)yi * (size_t)W_ + (size_t)xi;
#ifdef HAVE_TDM
            #pragma unroll 16
            for (int c = 0; c < C_; ++c) {
                out[obase + (size_t)c * (size_t)HW_] = tile[c * rem + tid];
            }
#else
            // Fallback: direct coalesced global reads (no TDM builtin).
            #pragma unroll 16
            for (int c = 0; c < C_; ++c) {
                out[obase + (size_t)c * (size_t)HW_] = src[(size_t)c * (size_t)P + (size_t)tid];
            }
#endif
        }
    }
}

// ---------------------------------------------------------------------------
// Launcher
// ---------------------------------------------------------------------------
extern "C" void kernel_launch(void* const* d_in, const int* in_sizes, int n_in,
                              void* d_out, int out_size, void* d_ws, size_t ws_size,
                              hipStream_t stream) {
    const float* x    = (const float*)d_in[0];
    const int*   inds = (const int*)d_in[1];
    float*       out  = (float*)d_out;

    const int P = in_sizes[0] / (B_ * C_);   // 12000

    // 1) zero-fill the output (must complete before the scatter; same stream).
    const int n4   = out_size / 4;
    const int tail = out_size - n4 * 4;
    {
        const int threads = 256;
        const int blocks  = (n4 + threads - 1) / threads;
        PPScatter_zero_kernel<<<blocks, threads, 0, stream>>>(
            (float4*)out, n4, out + (size_t)n4 * 4, tail);
    }

    // 2) TDM-staged scatter.
    const int bpb = (P + TILE - 1) / TILE;
    PPScatter_scatter_kernel<<<B_ * bpb, TILE, 0, stream>>>(x, inds, out, P, bpb);
}